// GATAttention_62818191671913
// MI455X (gfx1250) — compile-verified
//
#include <hip/hip_runtime.h>
#include <math.h>

typedef __attribute__((ext_vector_type(2))) float v2f;
typedef __attribute__((ext_vector_type(8))) float v8f;

// ---------------------------------------------------------------------------
// K0: zero scratch accumulators
// ---------------------------------------------------------------------------
__global__ void k_zero(float* __restrict__ p, int n) {
    int i = blockIdx.x * blockDim.x + threadIdx.x;
    if (i < n) p[i] = 0.f;
}

// ---------------------------------------------------------------------------
// K1: pack att_w (d x H) and gsl1_w (d x G) into Wc layout [k/4][n][k%4],
//     n in [0,32): cols 0..H-1 = att_w, cols H..H+G-1 = gsl1_w, rest 0.
// ---------------------------------------------------------------------------
__global__ void k_prep(const float* __restrict__ att_w, const float* __restrict__ gsl1_w,
                       float* __restrict__ wc, int d, int H, int G) {
    int total = d * 32;
    for (int idx = blockIdx.x * blockDim.x + threadIdx.x; idx < total;
         idx += gridDim.x * blockDim.x) {
        int c = idx >> 5;      // k index 0..d-1
        int n = idx & 31;      // packed column
        float v = 0.f;
        if (n < H)           v = att_w[c * H + n];
        else if (n < H + G)  v = gsl1_w[c * G + (n - H)];
        wc[((c >> 2) * 32 + n) * 4 + (c & 3)] = v;
    }
}

// ---------------------------------------------------------------------------
// K2: WMMA pass. Per wave: 16 rows of h (16x256) times Wc (256x32) via
//     V_WMMA_F32_16X16X4_F32, two 16x16 output tiles, 64 K-chunks.
//     Epilogue: A = relu(tile cols 0..7 + att_b), S = sigmoid(gsl2 over 16
//     hidden units spread across lanes, reduced with shfl_xor).
// ---------------------------------------------------------------------------
__global__ void k_pass1(const float* __restrict__ h, const float* __restrict__ wc,
                        const float* __restrict__ att_b, const float* __restrict__ gsl1_b,
                        const float* __restrict__ gsl2_w, const float* __restrict__ gsl2_b,
                        float* __restrict__ A_out, float* __restrict__ S_out,
                        int M, int blocks_per_batch) {
    __shared__ float wlds[8192];   // 256 x 32 packed weights, 32 KB
    const int tid = threadIdx.x;
    for (int i = tid; i < 8192; i += 256) wlds[i] = wc[i];
    __syncthreads();

    const int b    = blockIdx.x / blocks_per_batch;
    const int mblk = blockIdx.x % blocks_per_batch;
    const int wave = tid >> 5;
    const int lane = tid & 31;
    const int half = lane >> 4;    // 0: K kk..kk+1, 1: K kk+2..kk+3
    const int r    = lane & 15;    // A-matrix row / B-matrix column
    const int m0   = mblk * 128 + wave * 16;

    int lrow = m0 + r;             // row this lane loads for the A operand
    if (lrow > M - 1) lrow = M - 1;
    const float* hrow = h + ((size_t)b * M + lrow) * 256;
    const int koff = half * 2;

    v8f c0 = {};   // output tile cols 0..15
    v8f c1 = {};   // output tile cols 16..31
    for (int kk = 0; kk < 256; kk += 4) {
        // A 16x4 f32: lane holds h[row][kk+koff], h[row][kk+koff+1]
        v2f a = *(const v2f*)(hrow + kk + koff);
        const int chunk = kk >> 2;
        // B 4x16 f32: lane holds Wc[kk+koff..+1][n] for n=r (tile0), n=16+r (tile1)
        v2f b0 = *(const v2f*)(&wlds[(chunk * 32 + r) * 4 + koff]);
        v2f b1 = *(const v2f*)(&wlds[(chunk * 32 + 16 + r) * 4 + koff]);
        c0 = __builtin_amdgcn_wmma_f32_16x16x4_f32(false, a, false, b0, (short)0, c0, false, false);
        c1 = __builtin_amdgcn_wmma_f32_16x16x4_f32(false, a, false, b1, (short)0, c1, false, false);
    }

    // Epilogue. C/D layout: VGPR j, lanes 0-15 -> M=j, lanes 16-31 -> M=j+8, N=lane&15.
    const float ab  = (r < 8) ? att_b[r] : 0.f;
    const int   g   = (r < 8) ? (r + 8) : (r - 8);   // gsl hidden unit owned by this lane
    const float g1b = gsl1_b[g];
    const float g2w = gsl2_w[g];
    const float g2b = gsl2_b[0];

    #pragma unroll
    for (int j = 0; j < 8; ++j) {
        const int   mj  = m0 + half * 8 + j;
        const float cv0 = c0[j];
        const float cv1 = c1[j];
        // gsl2 contraction: lanes r<8 hold hidden g=r+8 in tile1, lanes r>=8 hold g=r-8 in tile0
        float contrib = fmaxf(((r < 8) ? cv1 : cv0) + g1b, 0.f) * g2w;
        contrib += __shfl_xor(contrib, 1);
        contrib += __shfl_xor(contrib, 2);
        contrib += __shfl_xor(contrib, 4);
        contrib += __shfl_xor(contrib, 8);
        if (mj < M) {
            if (r < 8) A_out[((size_t)b * M + mj) * 8 + r] = fmaxf(cv0 + ab, 0.f);
            if (r == 0) S_out[(size_t)b * M + mj] = 1.f / (1.f + expf(-(contrib + g2b)));
        }
    }
}

// ---------------------------------------------------------------------------
// K3: exact k-th largest of S per batch, radix select on float bit patterns
//     (all S > 0, so float order == unsigned-int order).
// ---------------------------------------------------------------------------
__global__ void k_select(const float* __restrict__ S, const int* __restrict__ kptr,
                         float* __restrict__ kth, int M) {
    __shared__ unsigned hist[256];
    __shared__ unsigned sh_sel, sh_rem;
    const int b = blockIdx.x;
    const unsigned* sb = (const unsigned*)(S + (size_t)b * M);
    int k = *kptr;
    if (k > M) k = M;
    if (k < 1) k = 1;

    unsigned prefix = 0;
    unsigned remaining = (unsigned)k;
    for (int pass = 3; pass >= 0; --pass) {
        const int shift = pass * 8;
        for (int i = threadIdx.x; i < 256; i += blockDim.x) hist[i] = 0;
        __syncthreads();
        for (int m = threadIdx.x; m < M; m += blockDim.x) {
            unsigned bits = sb[m];
            bool match = (pass == 3) || ((bits >> (shift + 8)) == (prefix >> (shift + 8)));
            if (match) atomicAdd(&hist[(bits >> shift) & 255u], 1u);
        }
        __syncthreads();
        if (threadIdx.x == 0) {
            unsigned rem = remaining, sel = 0;
            for (int v = 255; v >= 0; --v) {
                unsigned c = hist[v];
                if (c >= rem) { sel = (unsigned)v; break; }
                rem -= c;
            }
            sh_sel = sel;
            sh_rem = rem;
        }
        __syncthreads();
        prefix |= (sh_sel << shift);
        remaining = sh_rem;
        __syncthreads();
    }
    if (threadIdx.x == 0) kth[b] = __uint_as_float(prefix);
}

// ---------------------------------------------------------------------------
// K4: second pass over h. Thread owns column c=tid. Per 32-row group: stage
//     e = exp(A*mask) in LDS (8 heads x 32 rows), then every thread does
//     8 FMAs per h element. Also accumulates softmax denominators and the
//     per-column sum of h (for mean(h)). Writes the mask output.
// ---------------------------------------------------------------------------
__global__ void k_pass3(const float* __restrict__ h, const float* __restrict__ A,
                        const float* __restrict__ S, const float* __restrict__ kth,
                        float* __restrict__ mask_out, float* __restrict__ denom,
                        float* __restrict__ t_acc, float* __restrict__ hsum,
                        int M, int chunks) {
    const int ROWS = 512, GROUPS = 16;
    __shared__ float el[32][8];
    __shared__ float red[256];
    const int b  = blockIdx.x / chunks;
    const int ch = blockIdx.x % chunks;
    const int m0 = ch * ROWS;
    const int tid = threadIdx.x;
    const int rr = tid >> 3, hh = tid & 7;
    const float kv = kth[b];

    float dsum = 0.f, hs = 0.f;
    float tacc[8] = {0.f, 0.f, 0.f, 0.f, 0.f, 0.f, 0.f, 0.f};

    for (int g = 0; g < GROUPS; ++g) {
        const int mbase = m0 + g * 32;
        // phase A: thread (rr,hh) computes e for row mbase+rr, head hh
        {
            const int m = mbase + rr;
            float e = 0.f;
            if (m < M) {
                float s  = S[(size_t)b * M + m];
                float mk = (s >= kv) ? 1.f : 0.f;
                float a  = A[((size_t)b * M + m) * 8 + hh];
                e = expf(a * mk);   // mask==0 -> exp(0)=1, matching reference
                if (hh == 0) mask_out[(size_t)b * M + m] = mk;
            }
            el[rr][hh] = e;
            dsum += e;
        }
        __syncthreads();
        // phase B: column-parallel weighted accumulation
        {
            const float* hp = h + ((size_t)b * M + mbase) * 256 + tid;
            int rmax = M - mbase;
            if (rmax > 32) rmax = 32;
            for (int r2 = 0; r2 < rmax; ++r2) {
                float hv = hp[(size_t)r2 * 256];
                hs += hv;
                #pragma unroll
                for (int q = 0; q < 8; ++q) tacc[q] += el[r2][q] * hv;
            }
        }
        __syncthreads();
    }

    // reduce denominators (threads sharing hh = tid&7)
    red[tid] = dsum;
    __syncthreads();
    for (int off = 128; off >= 8; off >>= 1) {
        if (tid < off) red[tid] += red[tid + off];
        __syncthreads();
    }
    if (tid < 8) atomicAdd(&denom[b * 8 + tid], red[tid]);

    #pragma unroll
    for (int q = 0; q < 8; ++q)
        atomicAdd(&t_acc[((size_t)b * 8 + q) * 256 + tid], tacc[q]);
    atomicAdd(&hsum[b * 256 + tid], hs);
}

// ---------------------------------------------------------------------------
// K5: per-batch finish: out = normalize(relu(t@w_w/denom + relu(mean_h@res_w+b)))
// ---------------------------------------------------------------------------
__global__ void k_final(const float* __restrict__ t_acc, const float* __restrict__ denom,
                        const float* __restrict__ hsum, const float* __restrict__ w_w,
                        const float* __restrict__ res_w, const float* __restrict__ res_b,
                        float* __restrict__ out, int M) {
    __shared__ float red[256];
    __shared__ float meanv, varv;
    const int b = blockIdx.x, tid = threadIdx.x;
    const int hi = tid >> 5;                 // head for output column tid (32 cols/head)
    const float* tp = t_acc + ((size_t)b * 8 + hi) * 256;
    const float invM = 1.f / (float)M;

    float acc = 0.f, q = 0.f;
    for (int c = 0; c < 256; ++c) {
        acc += tp[c] * w_w[c * 256 + tid];
        q   += hsum[b * 256 + c] * res_w[c * 256 + tid];
    }
    float attn = acc / denom[b * 8 + hi];
    float qres = fmaxf(q * invM + res_b[tid], 0.f);
    float val  = fmaxf(attn + qres, 0.f);

    red[tid] = val;
    __syncthreads();
    for (int off = 128; off >= 1; off >>= 1) {
        if (tid < off) red[tid] += red[tid + off];
        __syncthreads();
    }
    if (tid == 0) meanv = red[0] * (1.f / 256.f);
    __syncthreads();
    float dx = val - meanv;
    red[tid] = dx * dx;
    __syncthreads();
    for (int off = 128; off >= 1; off >>= 1) {
        if (tid < off) red[tid] += red[tid + off];
        __syncthreads();
    }
    if (tid == 0) varv = red[0] * (1.f / 256.f);
    __syncthreads();
    out[(size_t)b * 256 + tid] = dx / sqrtf(varv + 1e-8f);
}

// ---------------------------------------------------------------------------
extern "C" void kernel_launch(void* const* d_in, const int* in_sizes, int n_in,
                              void* d_out, int out_size, void* d_ws, size_t ws_size,
                              hipStream_t stream) {
    const float* h      = (const float*)d_in[0];
    const float* att_w  = (const float*)d_in[1];
    const float* att_b  = (const float*)d_in[2];
    const float* w_w    = (const float*)d_in[3];
    const float* res_w  = (const float*)d_in[4];
    const float* res_b  = (const float*)d_in[5];
    const float* gsl1_w = (const float*)d_in[6];
    const float* gsl1_b = (const float*)d_in[7];
    const float* gsl2_w = (const float*)d_in[8];
    const float* gsl2_b = (const float*)d_in[9];
    const int*   kptr   = (const int*)d_in[10];

    const int d         = in_sizes[1] / in_sizes[2];            // 256
    const int H         = in_sizes[2];                          // 8 heads
    const int G         = in_sizes[7];                          // 16 gsl hidden
    const int out_units = in_sizes[5];                          // 256
    const long long BM  = (long long)in_sizes[0] / d;           // B*M
    const int B         = (int)(((long long)out_size - BM) / out_units);
    const int M         = (int)(BM / B);

    float* ws     = (float*)d_ws;
    float* A_ws   = ws;                                  // B*M*8
    float* S_ws   = A_ws + (size_t)B * M * 8;            // B*M
    float* Wc     = S_ws + (size_t)B * M;                // d*32
    float* kth    = Wc + (size_t)d * 32;                 // B
    float* denom  = kth + B;                             // B*8
    float* t_acc  = denom + (size_t)B * 8;               // B*8*256
    float* hs     = t_acc + (size_t)B * 8 * 256;         // B*256

    // zero the accumulators (denom, t_acc, hs are contiguous)
    const int nz = B * 8 + B * 8 * 256 + B * 256;
    k_zero<<<(nz + 255) / 256, 256, 0, stream>>>(denom, nz);

    k_prep<<<32, 256, 0, stream>>>(att_w, gsl1_w, Wc, d, H, G);

    const int bpb = (M + 127) / 128;
    k_pass1<<<B * bpb, 256, 0, stream>>>(h, Wc, att_b, gsl1_b, gsl2_w, gsl2_b,
                                         A_ws, S_ws, M, bpb);

    k_select<<<B, 256, 0, stream>>>(S_ws, kptr, kth, M);

    float* mask_out = (float*)d_out + (size_t)B * out_units;
    const int chunks = (M + 511) / 512;
    k_pass3<<<B * chunks, 256, 0, stream>>>(h, A_ws, S_ws, kth, mask_out,
                                            denom, t_acc, hs, M, chunks);

    k_final<<<B, 256, 0, stream>>>(t_acc, denom, hs, w_w, res_w, res_b,
                                   (float*)d_out, M);
}